// LocalContrastiveLoss_19636590477473
// MI455X (gfx1250) — compile-verified
//
#include <hip/hip_runtime.h>

// ---------------------------------------------------------------------------
// LocalContrastiveLoss for MI455X (gfx1250, wave32, WMMA).
// x: (16, 512, 62, 62) f32. Output: scalar f32 loss.
// Core trick: V_WMMA_F32_16X16X4_F32 with A rows = 16 channels' 4-elem patch
// vectors, B cols = partner vectors of the same channels -> diagonal of the
// 16x16 product = 16 per-channel dot products (K=4 matches the 2x2 patch).
// ---------------------------------------------------------------------------

typedef __attribute__((ext_vector_type(2))) float v2f;
typedef __attribute__((ext_vector_type(8))) float v8f;

#define LW        62          // W (row stride in the 62x62 image)
#define LHW       3844        // H*W
#define LC        512         // channels
#define TINV      10.0f       // 1/TEMP
#define LEPS      1e-6f

// POS_FLAT bases: y*62 + x for POS coords {0,30,60}^2
__constant__ int POSB[9] = { 0, 30, 60, 1860, 1890, 1920, 3720, 3750, 3780 };

// NEG bases per region (5 each), y*62 + x
__constant__ int NEGB[9][5] = {
    {   30, 1860, 1890,   60, 3720 },
    {    0,   60, 1860, 1890, 1920 },
    {    0,   30, 1890, 1920, 3780 },
    {    0,   30, 1890, 3720, 3750 },
    {    0,   30, 1860, 1920, 3750 },
    {   30,   60, 1890, 3750, 3780 },
    {    0, 1860, 1890, 3750, 3780 },
    { 1860, 1890, 1920, 3720, 3780 },
    {   60, 1890, 1920, 3720, 3750 }
};

// Per-channel dot of 4-vectors for 16 channels via one WMMA; extract diagonal.
// Diagonal layout (32-bit 16x16 C/D): n<8 -> lane n, vgpr n;
// n>=8 -> lane n+16, vgpr n-8.  => lanes 0-7 use sel=lane, lanes 24-31 sel=lane-24.
__device__ __forceinline__ float wdot(v2f a, v2f b, int sel, bool act) {
    v8f c = {};
    c = __builtin_amdgcn_wmma_f32_16x16x4_f32(
            /*neg_a=*/false, a, /*neg_b=*/false, b,
            /*c_mod=*/(short)0, c, /*reuse_a=*/false, /*reuse_b=*/false);
    float d = c[0];
    d = (sel == 1) ? c[1] : d;
    d = (sel == 2) ? c[2] : d;
    d = (sel == 3) ? c[3] : d;
    d = (sel == 4) ? c[4] : d;
    d = (sel == 5) ? c[5] : d;
    d = (sel == 6) ? c[6] : d;
    d = (sel == 7) ? c[7] : d;
    return act ? d : 0.0f;
}

__global__ __launch_bounds__(256)
void lcl_main(const float* __restrict__ x, float* __restrict__ part) {
    const int blk  = blockIdx.x;        // 0..71
    const int bp   = blk & 7;           // batch pair 0..7
    const int r    = blk >> 3;          // region 0..8
    const int tid  = threadIdx.x;
    const int lane = tid & 31;
    const int wave = tid >> 5;          // 0..7
    const int half = lane >> 4;         // which K-pair this lane carries
    const int cidx = lane & 15;         // channel within 16-tile
    const bool act = (lane < 8) || (lane >= 24);
    const int  sel = act ? ((lane < 8) ? lane : (lane - 24)) : 0;

    // element offsets for this lane's half of the 2x2 patch (k01 or k23)
    const int pb = POSB[r] + half * LW;
    int nb[5];
#pragma unroll
    for (int k = 0; k < 5; ++k) nb[k] = NEGB[r][k] + half * LW;

    float sS = 0.0f, sD1 = 0.0f, sD2 = 0.0f;

    // 32 tiles of 16 channels; wave w handles tiles w, w+8, w+16, w+24
    for (int t = wave; t < 32; t += 8) {
        const int ch = t * 16 + cidx;
        const float* __restrict__ p1 = x + ((size_t)(2 * bp)     * LC + ch) * LHW;
        const float* __restrict__ p2 = x + ((size_t)(2 * bp + 1) * LC + ch) * LHW;

        v2f P1 = { p1[pb], p1[pb + 1] };
        v2f P2 = { p2[pb], p2[pb + 1] };
        v2f N1[5], N2[5];
#pragma unroll
        for (int k = 0; k < 5; ++k) {
            N1[k] = (v2f){ p1[nb[k]], p1[nb[k] + 1] };
            N2[k] = (v2f){ p2[nb[k]], p2[nb[k] + 1] };
        }

        const float s11 = wdot(P1, P1, sel, act);
        const float s22 = wdot(P2, P2, sel, act);
        const float s12 = wdot(P1, P2, sel, act);
        const float r1 = __builtin_sqrtf(s11);
        const float r2 = __builtin_sqrtf(s22);

        sS += __expf(TINV * s12 / fmaxf(r1 * r2, LEPS));

#pragma unroll
        for (int k = 0; k < 5; ++k) {
            const float m1 = wdot(N1[k], N1[k], sel, act);
            const float m2 = wdot(N2[k], N2[k], sel, act);
            const float a1 = wdot(P1, N1[k], sel, act);   // p1 . n1k
            const float b1 = wdot(P1, N2[k], sel, act);   // p1 . n2k
            const float a2 = wdot(P2, N1[k], sel, act);   // p2 . n1k
            const float b2 = wdot(P2, N2[k], sel, act);   // p2 . n2k
            const float rn1 = __builtin_sqrtf(m1);
            const float rn2 = __builtin_sqrtf(m2);
            sD1 += __expf(TINV * a1 / fmaxf(r1 * rn1, LEPS))
                 + __expf(TINV * b1 / fmaxf(r1 * rn2, LEPS));
            sD2 += __expf(TINV * b2 / fmaxf(r2 * rn2, LEPS))
                 + __expf(TINV * a2 / fmaxf(r2 * rn1, LEPS));
        }
    }

    // Deterministic fixed-order block reduction (lanes 8..23 carry garbage -> mask)
    __shared__ float red0[256], red1[256], red2[256];
    red0[tid] = act ? sS  : 0.0f;
    red1[tid] = act ? sD1 : 0.0f;
    red2[tid] = act ? sD2 : 0.0f;
    __syncthreads();
    if (tid == 0) {
        float S = 0.0f, D1 = 0.0f, D2 = 0.0f;
        for (int i = 0; i < 256; ++i) { S += red0[i]; D1 += red1[i]; D2 += red2[i]; }
        // -log(S/(S+D1)) - log(S/(S+D2))
        part[blk] = __logf((S + D1) / S) + __logf((S + D2) / S);
    }
}

__global__ void lcl_final(const float* __restrict__ part, float* __restrict__ out) {
    if (threadIdx.x == 0 && blockIdx.x == 0) {
        float s = 0.0f;
        for (int i = 0; i < 72; ++i) s += part[i];   // fixed order -> deterministic
        out[0] = s * (1.0f / (16.0f * 9.0f));        // / BATCH / N_REGIONS
    }
}

extern "C" void kernel_launch(void* const* d_in, const int* in_sizes, int n_in,
                              void* d_out, int out_size, void* d_ws, size_t ws_size,
                              hipStream_t stream) {
    (void)in_sizes; (void)n_in; (void)out_size; (void)ws_size;
    const float* x = (const float*)d_in[0];
    float* part = (float*)d_ws;          // 72 floats of scratch
    lcl_main<<<72, 256, 0, stream>>>(x, part);
    lcl_final<<<1, 32, 0, stream>>>(part, (float*)d_out);
}